// SoftHausdorffLoss_46411416601084
// MI455X (gfx1250) — compile-verified
//
#include <hip/hip_runtime.h>

typedef __attribute__((ext_vector_type(2))) float v2f;
typedef __attribute__((ext_vector_type(8))) float v8f;
typedef __attribute__((ext_vector_type(4))) int   v4i;

// K2 = (ALPHA * log2(e))^2 ; WMMA produces K2*d^2 so t = exp2(-sqrt(c)) directly.
#define K2 52.034225765405935f
#define NB 4
#define NPTS 8192
#define ROWS_PER_WAVE 16
#define WAVES_PER_BLOCK 8
#define ROWS_PER_BLOCK (ROWS_PER_WAVE * WAVES_PER_BLOCK)   // 128
#define KTILE 16
#define NKT (NPTS / KTILE)                                 // 512
#define LDS_BYTES (NPTS * 4 + NPTS * 8)                    // lcol + lgt = 96 KB

#if __has_builtin(__builtin_amdgcn_global_load_async_to_lds_b128) && \
    __has_builtin(__builtin_amdgcn_s_wait_asynccnt)
#define USE_ASYNC_LDS 1
typedef __attribute__((address_space(1))) v4i ga_v4i;   // global (AS1)
typedef __attribute__((address_space(3))) v4i ls_v4i;   // LDS (AS3)
#endif

// ---------------------------------------------------------------------------
// Zero the column-sum accumulator region of the workspace.
// ---------------------------------------------------------------------------
__global__ void SoftHausdorff_zero(float* __restrict__ p, int n) {
    int i = blockIdx.x * blockDim.x + threadIdx.x;
    if (i < n) p[i] = 0.0f;
}

// ---------------------------------------------------------------------------
// Main kernel: pairwise scaled d^2 via V_WMMA_F32_16X16X4_F32 with embedding
//   A_m = K2*[1, -2px, -2py, |p|^2],  B_k = [|g|^2, gx, gy, 1]
//   -> A.B = K2*d^2 ;  t = exp(-alpha*d) = exp2(-sqrt(A.B))
// gt points staged in LDS once per block (async-to-LDS path when available);
// hot loop is ds_load + wmma + raw transcendentals, no VMEM.
// ---------------------------------------------------------------------------
__global__ __launch_bounds__(256) void SoftHausdorff_main(
        const float* __restrict__ pred, const float* __restrict__ gt,
        float* __restrict__ rowsum, float* __restrict__ colsum) {
    extern __shared__ float smem[];
    float*  lcol = smem;                         // NPTS floats   (32 KB)
    float2* lgt  = (float2*)(smem + NPTS);       // NPTS float2   (64 KB)

    const int batch = blockIdx.y;
    const int chunk = blockIdx.x;
    const int tid   = threadIdx.x;
    const int lane  = tid & 31;
    const int wave  = tid >> 5;
    const int half  = lane >> 4;   // which 16-lane half of the wave
    const int l15   = lane & 15;

    for (int i = tid; i < NPTS; i += blockDim.x) lcol[i] = 0.0f;

    // ---- Stage this batch's gt points (64 KB) into LDS ----
    {
        const float4* gsrc = (const float4*)(gt + (size_t)batch * NPTS * 2);
        float4*       ldst = (float4*)lgt;       // 4096 float4
#if USE_ASYNC_LDS
        for (int i = tid; i < NPTS / 2; i += 256)
            __builtin_amdgcn_global_load_async_to_lds_b128(
                (ga_v4i*)(gsrc + i), (ls_v4i*)(ldst + i), 0, 0);
        __builtin_amdgcn_s_wait_asynccnt(0);
#else
        for (int i = tid; i < NPTS / 2; i += 256)
            ldst[i] = gsrc[i];
#endif
    }
    __syncthreads();

    // ---- Build A fragment once: rows m0 .. m0+15, K2-scaled embedding ----
    const int m0 = chunk * ROWS_PER_BLOCK + wave * ROWS_PER_WAVE;
    const float2 p = ((const float2*)pred)[batch * NPTS + m0 + l15];
    const float e0 = K2;
    const float e1 = -2.0f * K2 * p.x;
    const float e2 = -2.0f * K2 * p.y;
    const float e3 = K2 * (p.x * p.x + p.y * p.y);
    // 32-bit A 16x4 layout: lanes 0-15 -> K=0,1 in VGPR0,1; lanes 16-31 -> K=2,3
    v2f a;
    a.x = half ? e2 : e0;
    a.y = half ? e3 : e1;

    float racc[8];
#pragma unroll
    for (int j = 0; j < 8; ++j) racc[j] = 0.0f;

#pragma unroll 4
    for (int kt = 0; kt < NKT; ++kt) {
        const float2 g = lgt[kt * KTILE + l15];  // ds_load_b64, no VMEM in loop
        const float f0 = g.x * g.x + g.y * g.y;
        const float f1 = g.x;
        const float f2 = g.y;
        const float f3 = 1.0f;
        // B 4x16 layout: N = lane&15; lanes 0-15 -> K=0,1; lanes 16-31 -> K=2,3
        v2f b;
        b.x = half ? f2 : f0;
        b.y = half ? f3 : f1;

        v8f c = {0.f, 0.f, 0.f, 0.f, 0.f, 0.f, 0.f, 0.f};
        // D = A x B + 0 : 16x16 tile of K2*d^2 in one WMMA
        c = __builtin_amdgcn_wmma_f32_16x16x4_f32(
                /*neg_a=*/false, a, /*neg_b=*/false, b,
                /*c_mod=*/(short)0, c, /*reuse_a=*/false, /*reuse_b=*/false);

        float csum = 0.0f;
#pragma unroll
        for (int j = 0; j < 8; ++j) {
            // v_sqrt_f32 with |abs| modifier, v_exp_f32 with -neg modifier
            float t = __builtin_amdgcn_exp2f(
                          -__builtin_amdgcn_sqrtf(__builtin_fabsf(c[j])));
            racc[j] += t;   // row partial (this lane's column only)
            csum    += t;   // column partial over this lane's 8 rows
        }
        // all 32 lanes add their 8-row partial; lanes L and L+16 share a column
        atomicAdd(&lcol[kt * KTILE + l15], csum);   // ds_add_f32, no exec branch
    }

    // ---- Row sums: reduce over the 16 lanes of each half (rows are per-VGPR) ----
#pragma unroll
    for (int j = 0; j < 8; ++j) {
        float r = racc[j];
        r += __shfl_xor(r, 1);
        r += __shfl_xor(r, 2);
        r += __shfl_xor(r, 4);
        r += __shfl_xor(r, 8);
        if (l15 == 0)
            rowsum[batch * NPTS + m0 + half * 8 + j] = r;
    }

    __syncthreads();
    // ---- Flush block-local column sums to global accumulator ----
    for (int i = tid; i < NPTS; i += blockDim.x)
        atomicAdd(&colsum[batch * NPTS + i], lcol[i]);
}

// ---------------------------------------------------------------------------
// Finalize: out = mean(-log rowsum) + mean(-log colsum), both means over 32768.
// rowsum and colsum are contiguous in the workspace -> single pass of 65536.
// ---------------------------------------------------------------------------
__global__ __launch_bounds__(256) void SoftHausdorff_finalize(
        const float* __restrict__ sums, float* __restrict__ out) {
    __shared__ float red[256];
    const int tid = threadIdx.x;
    float acc = 0.0f;
    for (int i = tid; i < 2 * NB * NPTS; i += 256)
        acc += __builtin_amdgcn_logf(sums[i]);   // raw v_log_f32 (base-2)
    red[tid] = acc;
    __syncthreads();
    for (int s = 128; s > 0; s >>= 1) {
        if (tid < s) red[tid] += red[tid + s];
        __syncthreads();
    }
    if (tid == 0)
        out[0] = red[0] * (-0.6931471805599453f / (NB * NPTS)); // -ln2 * mean
}

// ---------------------------------------------------------------------------
extern "C" void kernel_launch(void* const* d_in, const int* in_sizes, int n_in,
                              void* d_out, int out_size, void* d_ws, size_t ws_size,
                              hipStream_t stream) {
    (void)in_sizes; (void)n_in; (void)out_size; (void)ws_size;
    const float* pred = (const float*)d_in[0];
    const float* gt   = (const float*)d_in[1];
    float* rowsum = (float*)d_ws;               // NB*NPTS floats (fully overwritten)
    float* colsum = rowsum + NB * NPTS;         // NB*NPTS floats (atomic accum)
    float* out    = (float*)d_out;

    const int ncol = NB * NPTS;
    SoftHausdorff_zero<<<(ncol + 255) / 256, 256, 0, stream>>>(colsum, ncol);

    dim3 grid(NPTS / ROWS_PER_BLOCK, NB);       // (64, 4) blocks
    SoftHausdorff_main<<<grid, 256, LDS_BYTES, stream>>>(pred, gt, rowsum, colsum);

    SoftHausdorff_finalize<<<1, 256, 0, stream>>>(rowsum, out);
}